// MLP_1803886264477
// MI455X (gfx1250) — compile-verified
//
#include <hip/hip_runtime.h>

typedef __attribute__((ext_vector_type(2))) float v2f;
typedef __attribute__((ext_vector_type(4))) float v4f;
typedef __attribute__((ext_vector_type(8))) float v8f;

#define IN_DIM 784
#define HIDDEN 1024
#define OUT_DIM 10
#define BATCH   256

// ---------------------------------------------------------------------------
// Layer 1: h1[b][h] = relu( sum_i W1[h][i] * mask[h][i][b] * x[b][i] + b1[h] )
// mask is [H][I][B] (batch contiguous) -> stream it with 16B non-temporal
// loads. One block per h: 256 threads = 4 i-phases (iq) x 64 batch-quads (l).
// Each thread: v4f over batches 4l..4l+3, i = iq, iq+4, ... (196 iters).
// LDS reduction over the 4 i-phases at the end.
// ---------------------------------------------------------------------------
__global__ __launch_bounds__(256) void l1_dropconnect(
    const float* __restrict__ x, const float* __restrict__ mask,
    const float* __restrict__ W1, const float* __restrict__ b1,
    float* __restrict__ h1)
{
    const int h  = blockIdx.x;
    const int t  = threadIdx.x;
    const int iq = t >> 6;     // 0..3   i-phase
    const int l  = t & 63;     // 0..63  batch quad
    const int b0 = l << 2;     // first batch of this quad

    const v4f* __restrict__ mask4 =
        (const v4f*)(mask + (size_t)h * (IN_DIM * BATCH));
    const float* __restrict__ w1row = W1 + (size_t)h * IN_DIM;

    float ax = 0.f, ay = 0.f, az = 0.f, aw = 0.f;
    for (int i = iq; i < IN_DIM; i += 4) {
        // 16B/lane, lanes contiguous within a wave -> 512B contiguous bursts.
        v4f  m = __builtin_nontemporal_load(&mask4[i * 64 + l]);
        float w = w1row[i];
        ax += w * m.x * x[(b0 + 0) * IN_DIM + i];
        ay += w * m.y * x[(b0 + 1) * IN_DIM + i];
        az += w * m.z * x[(b0 + 2) * IN_DIM + i];
        aw += w * m.w * x[(b0 + 3) * IN_DIM + i];
    }

    __shared__ v4f red[4][64];
    v4f mine = {ax, ay, az, aw};
    red[iq][l] = mine;
    __syncthreads();

    if (iq == 0) {
        v4f r0 = red[0][l], r1 = red[1][l], r2 = red[2][l], r3 = red[3][l];
        float bias = b1[h];
        float s0 = r0.x + r1.x + r2.x + r3.x + bias;
        float s1 = r0.y + r1.y + r2.y + r3.y + bias;
        float s2 = r0.z + r1.z + r2.z + r3.z + bias;
        float s3 = r0.w + r1.w + r2.w + r3.w + bias;
        h1[(size_t)(b0 + 0) * HIDDEN + h] = s0 > 0.f ? s0 : 0.f;
        h1[(size_t)(b0 + 1) * HIDDEN + h] = s1 > 0.f ? s1 : 0.f;
        h1[(size_t)(b0 + 2) * HIDDEN + h] = s2 > 0.f ? s2 : 0.f;
        h1[(size_t)(b0 + 3) * HIDDEN + h] = s3 > 0.f ? s3 : 0.f;
    }
}

// ---------------------------------------------------------------------------
// Layer 2: h2 = relu(h1 @ W2^T + b2)   [256,1024] x [1024,1024]^T
// Full-precision f32 WMMA: V_WMMA_F32_16X16X4_F32, one wave per 16x16 tile.
// A fragment (16x4 f32): lane = M (l&15), VGPRs hold K = 2*half, 2*half+1.
// B fragment (4x16 f32): lane = N (l&15), VGPRs hold K = 2*half, 2*half+1.
// Both are a single aligned float2 load per K-step.
// ---------------------------------------------------------------------------
__global__ __launch_bounds__(256) void l2_gemm_wmma(
    const float* __restrict__ h1, const float* __restrict__ W2,
    const float* __restrict__ b2, float* __restrict__ h2)
{
    const int tile = blockIdx.x * 8 + (threadIdx.x >> 5);  // 1024 tiles
    const int lane = threadIdx.x & 31;
    const int tm   = tile & 15;   // batch tile (16 of them)
    const int to   = tile >> 4;   // output tile (64 of them)
    const int half = lane >> 4;
    const int r16  = lane & 15;

    const float* aPtr = h1 + (size_t)(tm * 16 + r16) * HIDDEN + 2 * half;
    const float* bPtr = W2 + (size_t)(to * 16 + r16) * HIDDEN + 2 * half;

    v8f c = {};
    for (int k0 = 0; k0 < HIDDEN; k0 += 4) {
        v2f a = *(const v2f*)(aPtr + k0);
        v2f b = *(const v2f*)(bPtr + k0);
        c = __builtin_amdgcn_wmma_f32_16x16x4_f32(
                false, a, false, b, (short)0, c, false, false);
    }

    const float bias = b2[to * 16 + r16];
    const int colD   = to * 16 + r16;
    #pragma unroll
    for (int r = 0; r < 8; ++r) {
        int row = tm * 16 + half * 8 + r;   // C/D layout: VGPR r -> M=r / M=8+r
        float v = c[r] + bias;
        h2[(size_t)row * HIDDEN + colD] = v > 0.f ? v : 0.f;
    }
}

// ---------------------------------------------------------------------------
// Layer 3: out = h2 @ W3^T + b3   [256,1024] x [10,1024]^T -> [256,10]
// Same WMMA tile; N-dim padded 10->16 by clamping the W3 row index (columns
// are independent in D, so garbage columns 10..15 are simply not stored).
// ---------------------------------------------------------------------------
__global__ __launch_bounds__(256) void l3_gemm_wmma(
    const float* __restrict__ h2, const float* __restrict__ W3,
    const float* __restrict__ b3, float* __restrict__ out)
{
    const int tile = blockIdx.x * 8 + (threadIdx.x >> 5);  // 16 tiles (tm)
    const int lane = threadIdx.x & 31;
    const int tm   = tile;
    const int half = lane >> 4;
    const int r16  = lane & 15;

    const int wrow = (r16 < OUT_DIM) ? r16 : (OUT_DIM - 1);  // clamp N
    const float* aPtr = h2 + (size_t)(tm * 16 + r16) * HIDDEN + 2 * half;
    const float* bPtr = W3 + (size_t)wrow * HIDDEN + 2 * half;

    v8f c = {};
    for (int k0 = 0; k0 < HIDDEN; k0 += 4) {
        v2f a = *(const v2f*)(aPtr + k0);
        v2f b = *(const v2f*)(bPtr + k0);
        c = __builtin_amdgcn_wmma_f32_16x16x4_f32(
                false, a, false, b, (short)0, c, false, false);
    }

    const float bias = (r16 < OUT_DIM) ? b3[r16] : 0.f;
    #pragma unroll
    for (int r = 0; r < 8; ++r) {
        int row = tm * 16 + half * 8 + r;
        if (r16 < OUT_DIM)
            out[(size_t)row * OUT_DIM + r16] = c[r] + bias;
    }
}

// ---------------------------------------------------------------------------
extern "C" void kernel_launch(void* const* d_in, const int* in_sizes, int n_in,
                              void* d_out, int out_size, void* d_ws, size_t ws_size,
                              hipStream_t stream) {
    const float* x    = (const float*)d_in[0];
    const float* mask = (const float*)d_in[1];
    const float* W1   = (const float*)d_in[2];
    const float* b1   = (const float*)d_in[3];
    const float* W2   = (const float*)d_in[4];
    const float* b2   = (const float*)d_in[5];
    const float* W3   = (const float*)d_in[6];
    const float* b3   = (const float*)d_in[7];
    float* out = (float*)d_out;

    float* h1 = (float*)d_ws;                       // 256*1024 f32 = 1 MB
    float* h2 = h1 + (size_t)BATCH * HIDDEN;        // 256*1024 f32 = 1 MB

    // Layer 1: 1024 blocks (one per hidden unit) x 256 threads.
    l1_dropconnect<<<HIDDEN, 256, 0, stream>>>(x, mask, W1, b1, h1);

    // Layer 2: 1024 tiles, 8 waves/block -> 128 blocks x 256 threads.
    l2_gemm_wmma<<<(BATCH / 16) * (HIDDEN / 16) / 8, 256, 0, stream>>>(h1, W2, b2, h2);

    // Layer 3: 16 tiles -> 2 blocks x 256 threads.
    l3_gemm_wmma<<<(BATCH / 16) / 8, 256, 0, stream>>>(h2, W3, b3, out);
}